// LengthTTVLossLayer_15496242004112
// MI455X (gfx1250) — compile-verified
//
#include <hip/hip_runtime.h>
#include <math.h>

typedef float v2f __attribute__((ext_vector_type(2)));
typedef float v4f __attribute__((ext_vector_type(4)));
typedef float v8f __attribute__((ext_vector_type(8)));

#define B_    16
#define T_    128
#define C_    8
#define HWC   25088   // 56*56*8
#define COLS4 6272    // HWC/4
#define TPB   128
#define NB    49      // COLS4 / TPB
#define NBP   52      // NB padded to multiple of 4 (WMMA K-chunks)

// ws layout (floats):
//   part_act: [8][16][NBP]  = 6656
//   part_tot: [8][16][NBP]  = 6656
//   part_ttv: [16][NBP]     = 832
#define OFF_TOT 6656
#define OFF_TTV 13312

// ---------------------------------------------------------------------------
// Pass 1: stream 205.5 MB once. One thread owns one float4 column of (h,w,c)
// and walks T, keeping the previous masked value in a register (TTV needs
// the t-1 neighbor; this avoids loading anything twice). Non-temporal 128-bit
// loads: data is single-use and larger than the 192 MB L2.
// ---------------------------------------------------------------------------
__global__ __launch_bounds__(TPB) void loss_pass1(const float* __restrict__ cam,
                                                  const int*   __restrict__ length,
                                                  float*       __restrict__ ws) {
    const int tid  = threadIdx.x;
    const int blk  = blockIdx.x;   // 0..NB-1
    const int b    = blockIdx.y;   // 0..15
    const int len  = length[b];
    const int col4 = blk * TPB + tid;

    const v4f* pv = (const v4f*)cam + (size_t)b * T_ * COLS4 + col4;

    v4f act = {0.f, 0.f, 0.f, 0.f};
    v4f tot = {0.f, 0.f, 0.f, 0.f};
    float ttv = 0.f;

    // t = 0 (length >= 1 always, so mask(t=0) == 1)
    v4f x = __builtin_nontemporal_load(pv);
    v4f prev = x;
    tot += x;
    act += x;

    for (int t = 1; t < T_; ++t) {
        x = __builtin_nontemporal_load(pv + (size_t)t * COLS4);
        tot += x;
        float m = (t < len) ? 1.0f : 0.0f;
        v4f a = x * m;
        act += a;
        v4f d = a - prev;
        ttv += fabsf(d.x) + fabsf(d.y) + fabsf(d.z) + fabsf(d.w);
        prev = a;
    }

    // Block reduction. Even threads carry channels 0-3, odd threads 4-7
    // (float4 spans half of C=8). Even strides (64..2) preserve parity, so
    // slot 0 ends with channels 0-3, slot 1 with channels 4-7.
    __shared__ float s_act[4][TPB];
    __shared__ float s_tot[4][TPB];
    __shared__ float s_ttv[TPB];
    s_act[0][tid] = act.x; s_act[1][tid] = act.y; s_act[2][tid] = act.z; s_act[3][tid] = act.w;
    s_tot[0][tid] = tot.x; s_tot[1][tid] = tot.y; s_tot[2][tid] = tot.z; s_tot[3][tid] = tot.w;
    s_ttv[tid] = ttv;
    __syncthreads();

    for (int s = TPB / 2; s >= 2; s >>= 1) {
        if (tid < s) {
#pragma unroll
            for (int i = 0; i < 4; ++i) {
                s_act[i][tid] += s_act[i][tid + s];
                s_tot[i][tid] += s_tot[i][tid + s];
            }
            s_ttv[tid] += s_ttv[tid + s];
        }
        __syncthreads();
    }

    if (tid == 0) {
#pragma unroll
        for (int i = 0; i < 4; ++i) {
            ws[((i    ) * B_ + b) * NBP + blk] = s_act[i][0];            // c = i
            ws[((4 + i) * B_ + b) * NBP + blk] = s_act[i][1];            // c = 4+i
            ws[OFF_TOT + ((i    ) * B_ + b) * NBP + blk] = s_tot[i][0];
            ws[OFF_TOT + ((4 + i) * B_ + b) * NBP + blk] = s_tot[i][1];
        }
        ws[OFF_TTV + b * NBP + blk] = s_ttv[0] + s_ttv[1];
    }
    // Zero the K-padding slots (blk 49..51) so pass-2 WMMA chunks of 4 are exact.
    if (blk == 0 && tid < (NBP - NB)) {
        int p = NB + tid;
#pragma unroll
        for (int c = 0; c < C_; ++c) {
            ws[(c * B_ + b) * NBP + p] = 0.f;
            ws[OFF_TOT + (c * B_ + b) * NBP + p] = 0.f;
        }
        ws[OFF_TTV + b * NBP + p] = 0.f;
    }
}

// ---------------------------------------------------------------------------
// Pass 2: single wave32. Cross-block accumulation done with
// V_WMMA_F32_16X16X4_F32 (fp32-exact): A = 16 batches x K=4 partials,
// B = ones(4x16)  =>  D[b][*] += sum_k A[b][k].
// A layout (ISA 7.12.2): lanes 0-15 hold {K=0,K=1}, lanes 16-31 hold {K=2,K=3}.
// D column 0: lane 0 has M=0..7 in v[0..7], lane 16 has M=8..15.
// ---------------------------------------------------------------------------
__global__ __launch_bounds__(32) void loss_pass2(const float* __restrict__ count,
                                                 const float* __restrict__ ws,
                                                 float*       __restrict__ out) {
    const int lane = threadIdx.x;
    const int row  = lane & 15;
    const int hi   = lane >> 4;

    const v2f bones = {1.0f, 1.0f};

    __shared__ float l_act[B_ * C_];
    __shared__ float l_tot[B_ * C_];
    __shared__ float l_ttv[2];

    // Global TTV sum: rows are b, K runs over the NBP block partials.
    {
        v8f acc = {};
        const float* src = ws + OFF_TTV;
        for (int j = 0; j < NBP; j += 4) {
            v2f a;
            int idx = row * NBP + j + hi * 2;
            a.x = src[idx];
            a.y = src[idx + 1];
            acc = __builtin_amdgcn_wmma_f32_16x16x4_f32(
                false, a, false, bones, (short)0, acc, false, false);
        }
        if (lane == 0 || lane == 16) {
            float s = 0.f;
#pragma unroll
            for (int m = 0; m < 8; ++m) s += acc[m];
            l_ttv[hi] = s;
        }
    }

    // Per-(b,c) active/total sums via WMMA, one channel at a time.
#pragma unroll 1
    for (int c = 0; c < C_; ++c) {
        v8f accA = {};
        v8f accT = {};
        const float* sa = ws + (c * B_) * NBP;
        const float* st = ws + OFF_TOT + (c * B_) * NBP;
        for (int j = 0; j < NBP; j += 4) {
            int idx = row * NBP + j + hi * 2;
            v2f a, t;
            a.x = sa[idx]; a.y = sa[idx + 1];
            t.x = st[idx]; t.y = st[idx + 1];
            accA = __builtin_amdgcn_wmma_f32_16x16x4_f32(
                false, a, false, bones, (short)0, accA, false, false);
            accT = __builtin_amdgcn_wmma_f32_16x16x4_f32(
                false, t, false, bones, (short)0, accT, false, false);
        }
        if (lane == 0 || lane == 16) {
#pragma unroll
            for (int m = 0; m < 8; ++m) {
                l_act[(hi * 8 + m) * C_ + c] = accA[m];
                l_tot[(hi * 8 + m) * C_ + c] = accT[m];
            }
        }
    }
    __syncthreads();

    // Huber + combine. Lane b handles batch b.
    if (lane < B_) {
        const int b = lane;
        float ah = 0.f, bh = 0.f;
#pragma unroll
        for (int c = 0; c < C_; ++c) {
            float as = l_act[b * C_ + c];
            float bs = l_tot[b * C_ + c] - as;   // blank = total - active
            float e  = as - count[b * C_ + c];
            float ae = fabsf(e);
            ah += (ae <= 1.0f) ? 0.5f * e * e : (ae - 0.5f);
            float ab = fabsf(bs);
            bh += (ab <= 1.0f) ? 0.5f * bs * bs : (ab - 0.5f);
        }
        float ttv = (l_ttv[0] + l_ttv[1]) * 0.1f;
        out[b] = ah * 0.125f + bh * 0.125f + ttv;
    }
}

extern "C" void kernel_launch(void* const* d_in, const int* in_sizes, int n_in,
                              void* d_out, int out_size, void* d_ws, size_t ws_size,
                              hipStream_t stream) {
    (void)in_sizes; (void)n_in; (void)out_size; (void)ws_size;
    const float* cam    = (const float*)d_in[0];
    const float* count  = (const float*)d_in[1];
    const int*   length = (const int*)d_in[2];
    float* out = (float*)d_out;
    float* ws  = (float*)d_ws;   // needs 14144 floats (~55 KB)

    dim3 grid1(NB, B_);
    loss_pass1<<<grid1, TPB, 0, stream>>>(cam, length, ws);
    loss_pass2<<<1, 32, 0, stream>>>(count, ws, out);
}